// GNNBlock_7851200217415
// MI455X (gfx1250) — compile-verified
//
#include <hip/hip_runtime.h>

typedef __attribute__((ext_vector_type(2))) float v2f;
typedef __attribute__((ext_vector_type(8))) float v8f;

#define N_NODES 100000
#define N_EDGES 1600000
#define D 128
#define LN_EPS 1e-5f

// ---------------------------------------------------------------------------
// Kernel 1: LayerNorm (affine) + ReLU. One wave per node, 4 floats per lane.
// ---------------------------------------------------------------------------
__global__ void __launch_bounds__(256) ln_relu_kernel(
    const float* __restrict__ x, const float* __restrict__ gamma,
    const float* __restrict__ beta, float* __restrict__ h) {
  const int wave = threadIdx.x >> 5;
  const int lane = threadIdx.x & 31;
  const int node = blockIdx.x * 8 + wave;   // grid = N_NODES/8 exactly

  float4 v = ((const float4*)(x + (size_t)node * D))[lane];
  float s = v.x + v.y + v.z + v.w;
#pragma unroll
  for (int off = 16; off >= 1; off >>= 1) s += __shfl_xor(s, off, 32);
  const float mu = s * (1.0f / D);

  const float dx = v.x - mu, dy = v.y - mu, dz = v.z - mu, dw = v.w - mu;
  float sq = dx * dx + dy * dy + dz * dz + dw * dw;
#pragma unroll
  for (int off = 16; off >= 1; off >>= 1) sq += __shfl_xor(sq, off, 32);
  const float rstd = rsqrtf(sq * (1.0f / D) + LN_EPS);

  float4 g = ((const float4*)gamma)[lane];
  float4 b = ((const float4*)beta)[lane];
  float4 o;
  o.x = fmaxf(dx * rstd * g.x + b.x, 0.0f);
  o.y = fmaxf(dy * rstd * g.y + b.y, 0.0f);
  o.z = fmaxf(dz * rstd * g.z + b.z, 0.0f);
  o.w = fmaxf(dw * rstd * g.w + b.w, 0.0f);
  ((float4*)(h + (size_t)node * D))[lane] = o;
}

// ---------------------------------------------------------------------------
// Kernel 2: zero scratch (msg_sum + deg, contiguous), float4 stores.
// ---------------------------------------------------------------------------
__global__ void zero_kernel(float4* __restrict__ p, long long n4) {
  long long i = (long long)blockIdx.x * blockDim.x + threadIdx.x;
  if (i < n4) p[i] = make_float4(0.f, 0.f, 0.f, 0.f);
}

// ---------------------------------------------------------------------------
// Kernel 3: edge scatter. One wave per edge; lane handles 4 features.
// f32 global atomics land in L2 (msg_sum is L2-resident: 51 MB of 192 MB).
// ---------------------------------------------------------------------------
__global__ void __launch_bounds__(256) scatter_kernel(
    const long long* __restrict__ ei, const float* __restrict__ h,
    float* __restrict__ msg, float* __restrict__ deg) {
  const int wave = threadIdx.x >> 5;
  const int lane = threadIdx.x & 31;
  const long long e = (long long)blockIdx.x * 8 + wave;  // grid = N_EDGES/8

  const long long src = ei[e];
  const long long dst = ei[N_EDGES + e];

  float4 v = ((const float4*)(h + src * D))[lane];
  float* m = msg + dst * D + lane * 4;
  atomicAdd(m + 0, v.x);
  atomicAdd(m + 1, v.y);
  atomicAdd(m + 2, v.z);
  atomicAdd(m + 3, v.w);
  if (lane == 0) atomicAdd(deg + dst, 1.0f);
}

// ---------------------------------------------------------------------------
// Kernel 4: out = (msg/deg) @ Wl^T + bl + h @ Wr^T via V_WMMA_F32_16X16X4_F32.
// Block = 256 threads (8 waves) = one 16-row M-tile, wave w = 16 N-columns.
// A tiles staged in LDS (degree scaling fused); B read straight from the
// L2-hot 128KB of weights. 64 f32 WMMAs per wave.
// ---------------------------------------------------------------------------
__global__ void __launch_bounds__(256) sage_gemm_kernel(
    const float* __restrict__ h, const float* __restrict__ msg,
    const float* __restrict__ deg, const float* __restrict__ Wl,
    const float* __restrict__ bl, const float* __restrict__ Wr,
    float* __restrict__ out) {
  __shared__ float lds_h[16 * D];
  __shared__ float lds_m[16 * D];

  const int t = threadIdx.x;
  const int mbase = blockIdx.x * 16;  // grid = N_NODES/16 exactly

  // Cooperative stage-in: 512 float4 per matrix, 2 per thread.
#pragma unroll
  for (int i = 0; i < 2; ++i) {
    const int idx4 = t + i * 256;       // 0..511
    const int row = idx4 >> 5;          // 32 float4 per row
    float4 hv = ((const float4*)(h + (size_t)(mbase + row) * D))[idx4 & 31];
    ((float4*)lds_h)[idx4] = hv;
    const float r = 1.0f / fmaxf(deg[mbase + row], 1.0f);
    float4 mv = ((const float4*)(msg + (size_t)(mbase + row) * D))[idx4 & 31];
    mv.x *= r; mv.y *= r; mv.z *= r; mv.w *= r;
    ((float4*)lds_m)[idx4] = mv;
  }
  __syncthreads();

  const int wave = t >> 5;
  const int lane = t & 31;
  const int arow = lane & 15;           // A: M = lane%16
  const int krow = (lane >> 4) * 2;     // A/B: K sub-offset = 2*(lane/16)
  const int ncol = wave * 16 + arow;    // B/D: N = 16*wave + lane%16

  v8f c = {};
  // h @ Wr^T : B[k][n] = Wr[n][k], contiguous pair -> 64-bit loads
#pragma unroll
  for (int kb = 0; kb < D; kb += 4) {
    v2f a = *(const v2f*)(lds_h + arow * D + kb + krow);
    v2f b = *(const v2f*)(Wr + (size_t)ncol * D + kb + krow);
    c = __builtin_amdgcn_wmma_f32_16x16x4_f32(false, a, false, b, (short)0, c,
                                              false, false);
  }
  // mean_aggr @ Wl^T
#pragma unroll
  for (int kb = 0; kb < D; kb += 4) {
    v2f a = *(const v2f*)(lds_m + arow * D + kb + krow);
    v2f b = *(const v2f*)(Wl + (size_t)ncol * D + kb + krow);
    c = __builtin_amdgcn_wmma_f32_16x16x4_f32(false, a, false, b, (short)0, c,
                                              false, false);
  }

  const float bias = bl[ncol];
  const int rowhi = (lane >> 4) * 8;    // D: M = r + 8*(lane/16)
#pragma unroll
  for (int r = 0; r < 8; ++r)
    out[(size_t)(mbase + rowhi + r) * D + ncol] = c[r] + bias;
}

// ---------------------------------------------------------------------------
extern "C" void kernel_launch(void* const* d_in, const int* in_sizes, int n_in,
                              void* d_out, int out_size, void* d_ws,
                              size_t ws_size, hipStream_t stream) {
  const float* x = (const float*)d_in[0];
  const long long* ei = (const long long*)d_in[1];  // int64 edge_index
  const float* gamma = (const float*)d_in[2];
  const float* beta = (const float*)d_in[3];
  const float* Wl = (const float*)d_in[4];
  const float* bl = (const float*)d_in[5];
  const float* Wr = (const float*)d_in[6];
  float* out = (float*)d_out;

  float* h = (float*)d_ws;                          // N*D
  float* msg = h + (size_t)N_NODES * D;             // N*D
  float* dg = msg + (size_t)N_NODES * D;            // N

  ln_relu_kernel<<<N_NODES / 8, 256, 0, stream>>>(x, gamma, beta, h);

  const long long nz4 = ((long long)N_NODES * D + N_NODES) / 4;  // msg+deg
  zero_kernel<<<(int)((nz4 + 255) / 256), 256, 0, stream>>>((float4*)msg, nz4);

  scatter_kernel<<<N_EDGES / 8, 256, 0, stream>>>(ei, h, msg, dg);

  sage_gemm_kernel<<<N_NODES / 16, 256, 0, stream>>>(h, msg, dg, Wl, bl, Wr,
                                                     out);
}